// SOONet_36301063586212
// MI455X (gfx1250) — compile-verified
//
#include <hip/hip_runtime.h>
#include <hip/hip_bf16.h>

// ---------------------------------------------------------------------------
// SOONet-style pipeline for MI455X (gfx1250, wave32):
//   1) sorted-segment mean pooling: b128 non-temporal streaming, no atomics
//   2) query L2 normalization
//   3) cosine sim GEMM via V_WMMA_F32_16X16X4_F32 (exact fp32 matrix core)
//   4) stable top-8 per query
//   5) per-(q,k) bbox regression with wave32 shuffle reduction
// ---------------------------------------------------------------------------

#define HIDDEN    512
#define N_QUERIES 256
#define N_EVENTS  2048
#define TOPK      8
#define INV_FRAME_RATE 0.2f
#define NEG_BIG   (-1.0e30f)

typedef __attribute__((ext_vector_type(2))) float v2f;
typedef __attribute__((ext_vector_type(4))) float v4f;
typedef __attribute__((ext_vector_type(8))) float v8f;

// -------------------- 1) per-event pooling (events sorted) -----------------
// One block (256 thr) per event. Frames of an event are contiguous because
// video_events is sorted -> two binary searches give [lo, hi).
// 128 threads cover a 512-float row as float4 (b128 loads, 512B/wave32 req);
// the two 128-thread groups take even/odd frames and merge through LDS.
// vf is read exactly once and exceeds L2 (205MB > 192MB) -> non-temporal.
__global__ void event_pool_kernel(const float* __restrict__ vf,
                                  const int*   __restrict__ ev,
                                  int n_frames,
                                  float* __restrict__ eavg,   // [E, D] mean
                                  float* __restrict__ enrm,   // [E, D] normalized
                                  float* __restrict__ ets)    // [E, 2] timestamps
{
    __shared__ int   sb[2];
    __shared__ v4f   red4[256];
    __shared__ float red[128];
    __shared__ float sinv;

    const int e = blockIdx.x;
    const int t = threadIdx.x;

    if (t == 0) {
        int lo = 0, hi = n_frames;                 // lower_bound(e)
        while (lo < hi) { int m = (lo + hi) >> 1; if (ev[m] < e) lo = m + 1; else hi = m; }
        sb[0] = lo;
        int lo2 = lo, hi2 = n_frames;              // lower_bound(e+1)
        while (lo2 < hi2) { int m = (lo2 + hi2) >> 1; if (ev[m] < e + 1) lo2 = m + 1; else hi2 = m; }
        sb[1] = lo2;
    }
    __syncthreads();
    const int lo = sb[0], hi = sb[1];
    const int cnt = hi - lo;

    const int fq = t >> 7;          // frame-parity group: 0 or 1
    const int c  = t & 127;         // float4 column 0..127 (covers 512 dims)

    const v4f* vf4 = (const v4f*)vf;
    v4f acc = {0.0f, 0.0f, 0.0f, 0.0f};
    for (int i = lo + fq; i < hi; i += 2) {
        acc += __builtin_nontemporal_load(vf4 + (size_t)i * 128 + c);
    }
    red4[t] = acc;
    __syncthreads();

    const float invc = 1.0f / fmaxf((float)cnt, 1.0f);
    v4f tot = {0.0f, 0.0f, 0.0f, 0.0f};
    if (t < 128) {
        tot = red4[t] + red4[t + 128];
        tot *= invc;
        ((v4f*)eavg)[(size_t)e * 128 + t] = tot;
        red[t] = tot.x * tot.x + tot.y * tot.y + tot.z * tot.z + tot.w * tot.w;
    }
    __syncthreads();
    for (int off = 64; off > 0; off >>= 1) {
        if (t < off) red[t] += red[t + off];
        __syncthreads();
    }
    if (t == 0) sinv = 1.0f / fmaxf(sqrtf(red[0]), 1e-12f);
    __syncthreads();
    if (t < 128) {
        ((v4f*)enrm)[(size_t)e * 128 + t] = tot * sinv;
    }

    if (t == 0) {
        const int s  = (cnt > 0) ? lo       : 0;
        const int en = (cnt > 0) ? (hi - 1) : 0;
        ets[e * 2 + 0] = (float)s  * INV_FRAME_RATE;
        ets[e * 2 + 1] = (float)en * INV_FRAME_RATE;
    }
}

// -------------------- 2) query normalization -------------------------------
__global__ void qnorm_kernel(const float* __restrict__ qf,
                             float* __restrict__ qn)
{
    __shared__ float red[256];
    __shared__ float sinv;
    const int q = blockIdx.x;
    const int t = threadIdx.x;
    const float a0 = qf[(size_t)q * HIDDEN + t];
    const float a1 = qf[(size_t)q * HIDDEN + t + 256];
    red[t] = a0 * a0 + a1 * a1;
    __syncthreads();
    for (int off = 128; off > 0; off >>= 1) {
        if (t < off) red[t] += red[t + off];
        __syncthreads();
    }
    if (t == 0) sinv = 1.0f / fmaxf(sqrtf(red[0]), 1e-12f);
    __syncthreads();
    qn[(size_t)q * HIDDEN + t]       = a0 * sinv;
    qn[(size_t)q * HIDDEN + t + 256] = a1 * sinv;
}

// -------------------- 3) sim = qn @ en^T via fp32 WMMA ---------------------
// One wave32 per 16x16 output tile. K panels of 64 staged in LDS with row
// stride 66 (even -> 8B aligned, and (2m+k)%64 spreads across 16 banks).
#define LROW 66
__global__ void sim_wmma_kernel(const float* __restrict__ qn,
                                const float* __restrict__ en,
                                float* __restrict__ sim)    // [Q, E]
{
    __shared__ float lq[16 * LROW];
    __shared__ float le[16 * LROW];

    const int eb = blockIdx.x * 16;
    const int qb = blockIdx.y * 16;
    const int t  = threadIdx.x;          // 0..31

    v8f c = {};

    for (int kk = 0; kk < HIDDEN; kk += 64) {
        for (int idx = t; idx < 16 * 64; idx += 32) {
            const int r = idx >> 6, cc = idx & 63;
            lq[r * LROW + cc] = qn[(size_t)(qb + r) * HIDDEN + kk + cc];
            le[r * LROW + cc] = en[(size_t)(eb + r) * HIDDEN + kk + cc];
        }
        __syncthreads();

        const int m  = t & 15;
        const int h2 = (t >> 4) << 1;    // 0 (lanes 0-15: K 0,1) or 2 (lanes 16-31: K 2,3)
        #pragma unroll
        for (int k0 = 0; k0 < 64; k0 += 4) {
            v2f a, b;
            a.x = lq[m * LROW + k0 + h2];
            a.y = lq[m * LROW + k0 + h2 + 1];
            b.x = le[m * LROW + k0 + h2];
            b.y = le[m * LROW + k0 + h2 + 1];
            // D = A(16x4 f32) * B(4x16 f32) + C  -- exact fp32 matrix op
            c = __builtin_amdgcn_wmma_f32_16x16x4_f32(
                    false, a, false, b, (short)0, c, false, false);
        }
        __syncthreads();
    }

    // C/D layout: VGPR r -> M = r + 8*(lane>=16); N = lane&15
    const int nlane = t & 15;
    const int mofs  = (t >> 4) * 8;
    #pragma unroll
    for (int r = 0; r < 8; ++r) {
        const int M = r + mofs;
        sim[(size_t)(qb + M) * N_EVENTS + eb + nlane] = c[r];
    }
}

// -------------------- 4) stable top-8 per query ----------------------------
__global__ void topk_kernel(const float* __restrict__ sim,
                            int* __restrict__ topi)       // [Q, K]
{
    __shared__ float vals[N_EVENTS];
    __shared__ float rv[256];
    __shared__ int   ri[256];

    const int q = blockIdx.x;
    const int t = threadIdx.x;

    for (int i = t; i < N_EVENTS; i += 256)
        vals[i] = sim[(size_t)q * N_EVENTS + i];
    __syncthreads();

    for (int sel = 0; sel < TOPK; ++sel) {
        float bv = NEG_BIG - 1.0f;
        int   bi = 0x7fffffff;
        #pragma unroll
        for (int j = 0; j < N_EVENTS / 256; ++j) {
            const int i = t + j * 256;
            const float v = vals[i];
            if (v > bv || (v == bv && i < bi)) { bv = v; bi = i; }
        }
        rv[t] = bv; ri[t] = bi;
        __syncthreads();
        for (int off = 128; off > 0; off >>= 1) {
            if (t < off) {
                const float ov = rv[t + off]; const int oi = ri[t + off];
                if (ov > rv[t] || (ov == rv[t] && oi < ri[t])) { rv[t] = ov; ri[t] = oi; }
            }
            __syncthreads();
        }
        if (t == 0) {
            topi[q * TOPK + sel] = ri[0];
            vals[ri[0]] = NEG_BIG;       // mask out for next round
        }
        __syncthreads();
    }
}

// -------------------- 5) bbox regression, one wave32 per (q,k) -------------
__global__ void regress_kernel(const int*   __restrict__ topi,
                               const float* __restrict__ eavg,
                               const float* __restrict__ ets,
                               const float* __restrict__ qf,
                               const float* __restrict__ W,   // [D, 2]
                               const float* __restrict__ b,   // [2]
                               float* __restrict__ out)       // [Q, K, 2]
{
    const int q    = blockIdx.x;
    const int t    = threadIdx.x;
    const int k    = t >> 5;        // wave id 0..7 -> pick
    const int lane = t & 31;

    const int ei = topi[q * TOPK + k];

    float s0 = 0.0f, s1 = 0.0f;
    for (int d = lane; d < HIDDEN; d += 32) {
        const float f = eavg[(size_t)ei * HIDDEN + d] * qf[(size_t)q * HIDDEN + d];
        s0 += f * W[d * 2 + 0];
        s1 += f * W[d * 2 + 1];
    }
    #pragma unroll
    for (int off = 16; off > 0; off >>= 1) {
        s0 += __shfl_down(s0, off, 32);
        s1 += __shfl_down(s1, off, 32);
    }
    if (lane == 0) {
        const float t0  = ets[ei * 2 + 0];
        const float t1  = ets[ei * 2 + 1];
        const float dur = t1 - t0;
        out[(q * TOPK + k) * 2 + 0] = t0 + tanhf(s0 + b[0]) * dur;
        out[(q * TOPK + k) * 2 + 1] = t1 + tanhf(s1 + b[1]) * dur;
    }
}

// ---------------------------------------------------------------------------
extern "C" void kernel_launch(void* const* d_in, const int* in_sizes, int n_in,
                              void* d_out, int out_size, void* d_ws, size_t ws_size,
                              hipStream_t stream)
{
    const float* vf = (const float*)d_in[0];   // [1, N, 512]
    const float* qf = (const float*)d_in[1];   // [256, 512]
    const int*   ev = (const int*)  d_in[2];   // [N] sorted
    const float* W  = (const float*)d_in[3];   // [512, 2]
    const float* b  = (const float*)d_in[4];   // [2]
    float* out = (float*)d_out;                // [256, 8, 2]

    const int n_frames = in_sizes[2];

    // workspace layout (16B-aligned slabs)
    char* ws = (char*)d_ws;
    float* eavg = (float*)ws; ws += sizeof(float) * (size_t)N_EVENTS * HIDDEN;   // 4 MB
    float* enrm = (float*)ws; ws += sizeof(float) * (size_t)N_EVENTS * HIDDEN;   // 4 MB
    float* qn   = (float*)ws; ws += sizeof(float) * (size_t)N_QUERIES * HIDDEN;  // 0.5 MB
    float* sim  = (float*)ws; ws += sizeof(float) * (size_t)N_QUERIES * N_EVENTS;// 2 MB
    float* ets  = (float*)ws; ws += sizeof(float) * (size_t)N_EVENTS * 2;        // 16 KB
    int*   topi = (int*)ws;   ws += sizeof(int)   * (size_t)N_QUERIES * TOPK;    // 8 KB

    event_pool_kernel<<<N_EVENTS, 256, 0, stream>>>(vf, ev, n_frames, eavg, enrm, ets);
    qnorm_kernel<<<N_QUERIES, 256, 0, stream>>>(qf, qn);
    sim_wmma_kernel<<<dim3(N_EVENTS / 16, N_QUERIES / 16), 32, 0, stream>>>(qn, enrm, sim);
    topk_kernel<<<N_QUERIES, 256, 0, stream>>>(sim, topi);
    regress_kernel<<<N_QUERIES, 256, 0, stream>>>(topi, eavg, ets, qf, W, b, out);
}